// MarginLoss_10101763080262
// MI455X (gfx1250) — compile-verified
//
#include <hip/hip_runtime.h>
#include <math.h>

// ---------------------------------------------------------------------------
// D2-Net style hard-negative margin loss, fused for MI455X (gfx1250).
// B=2, C=64, H=W=64, N=4096.  One f32 WMMA GEMM (S = desc1^T desc2) feeds both
// masked-min reductions (dm1 = dm2^T), fused into the epilogue so the 64 MB
// distance matrices are never materialized.
// ---------------------------------------------------------------------------

typedef __attribute__((ext_vector_type(2))) float v2f;
typedef __attribute__((ext_vector_type(8))) float v8f;

#define B_    2
#define C_    64
#define HW_   64
#define N_    4096     // HW_*HW_
#define TILE_ 64

// Workspace layout (bytes). Total ~4.33 MB.
static constexpr size_t OFF_D1T = 0;                                   // [B][N][C] f32, 2 MB
static constexpr size_t OFF_D2T = OFF_D1T + (size_t)B_ * N_ * C_ * 4;  // 2 MB
static constexpr size_t OFF_POS = OFF_D2T + (size_t)B_ * N_ * C_ * 4;  // [B][N][2] i32, 64 KB
static constexpr size_t OFF_M1  = OFF_POS + (size_t)B_ * N_ * 2 * 4;   // [B][N] f32, 32 KB
static constexpr size_t OFF_M2  = OFF_M1  + (size_t)B_ * N_ * 4;       // [B][N] f32, 32 KB

// Float atomic max via sign-split int/uint atomics (monotone bit encodings).
__device__ __forceinline__ void atomicMaxF(float* addr, float val) {
  if (val >= 0.0f) atomicMax((int*)addr, __float_as_int(val));
  else             atomicMin((unsigned int*)addr, __float_as_uint(val));
}

// ---------------------------------------------------------------------------
// init: rounded warp coords, -inf the max accumulators, zero the outputs.
// ---------------------------------------------------------------------------
__global__ void init_kernel(const float* __restrict__ tc, int* __restrict__ pos2,
                            float* __restrict__ M1, float* __restrict__ M2,
                            float* __restrict__ out) {
  int idx = blockIdx.x * blockDim.x + threadIdx.x;
  if (idx < 3) out[idx] = 0.0f;
  if (idx >= B_ * N_) return;
  int b = idx >> 12, n = idx & (N_ - 1);
  const float* t = tc + (size_t)b * 2 * N_;
  int r = (int)rintf(t[n]);         // rintf = round-half-even, matches jnp.round
  int c = (int)rintf(t[N_ + n]);
  r = min(max(r, 0), HW_ - 1);
  c = min(max(c, 0), HW_ - 1);
  pos2[2 * idx + 0] = r;
  pos2[2 * idx + 1] = c;
  M1[idx] = -1.0e30f;
  M2[idx] = -1.0e30f;
}

// ---------------------------------------------------------------------------
// prep_desc: normalize descriptor columns (length C=64) and store transposed
// [N][C] so each descriptor is 256 contiguous bytes (WMMA fragments become
// single b64 lane loads). gather!=0 gathers source pixels through pos2.
// ---------------------------------------------------------------------------
__global__ void __launch_bounds__(64)
prep_desc(const float* __restrict__ img, const int* __restrict__ pos2,
          int gather, float* __restrict__ dT) {
  __shared__ float tile[64][65];  // +1 pad: conflict-free transpose
  __shared__ float invn[64];
  int b  = blockIdx.y;
  int n0 = blockIdx.x * 64;
  int t  = threadIdx.x;           // lane over pixel index for the read phase
  int srcn = n0 + t;
  if (gather) {
    size_t p = ((size_t)b * N_ + n0 + t) * 2;
    srcn = pos2[p] * HW_ + pos2[p + 1];
  }
  const float* src = img + (size_t)b * C_ * N_;
  float ss = 0.0f;
#pragma unroll 4
  for (int c = 0; c < C_; ++c) {
    float v = src[(size_t)c * N_ + srcn];
    tile[t][c] = v;
    ss += v * v;
  }
  invn[t] = 1.0f / fmaxf(sqrtf(ss), 1e-12f);
  __syncthreads();
  float* dst = dT + ((size_t)b * N_ + n0) * C_;
  for (int row = 0; row < 64; ++row)      // lane over channel: coalesced writes
    dst[(size_t)row * C_ + t] = tile[row][t] * invn[row];
}

// ---------------------------------------------------------------------------
// gemm_minmax: 64x64 tile of S = D1^T D2 via v_wmma_f32_16x16x4_f32, K=64.
// 4 waves / block, wave w owns rows [tI+16w, tI+16w+16) x all 64 cols
// (4 C/D fragments, 16 chained WMMAs each). Epilogue fuses both masked maxes.
// D fragment layout (ISA 7.12.2): vgpr r, lane<16 -> (M=r, N=lane),
// lane>=16 -> (M=8+r, N=lane-16).  A 16x4 f32: lane<16 holds K=0,1;
// lane>=16 holds K=2,3 (B mirrored).
// ---------------------------------------------------------------------------
__global__ void __launch_bounds__(128)
gemm_minmax_kernel(const float* __restrict__ d1T, const float* __restrict__ d2T,
                   const int* __restrict__ pos2,
                   float* __restrict__ M1, float* __restrict__ M2) {
  const int tiles = N_ / TILE_;                 // 64
  int b    = blockIdx.y;
  int tI   = (blockIdx.x / tiles) * TILE_;
  int tJ   = (blockIdx.x % tiles) * TILE_;
  int wave = threadIdx.x >> 5;
  int lane = threadIdx.x & 31;
  int half = lane >> 4;                         // 0: K lo / M lo, 1: K hi / M hi
  int l    = lane & 15;

  const float* D1 = d1T + (size_t)b * N_ * C_;
  const float* D2 = d2T + (size_t)b * N_ * C_;
  const int*   P2 = pos2 + (size_t)b * N_ * 2;

  int i0 = tI + wave * 16;
  const float* arow = D1 + (size_t)(i0 + l) * C_ + 2 * half;
  const float* brow[4];
#pragma unroll
  for (int f = 0; f < 4; ++f)
    brow[f] = D2 + (size_t)(tJ + 16 * f + l) * C_ + 2 * half;

  const v8f vzero = {0.f, 0.f, 0.f, 0.f, 0.f, 0.f, 0.f, 0.f};
  v8f acc[4] = {vzero, vzero, vzero, vzero};

#pragma unroll
  for (int k = 0; k < 16; ++k) {                // K = 4k .. 4k+3
    v2f a = *(const v2f*)(arow + 4 * k);        // A reused across 4 col tiles
#pragma unroll
    for (int f = 0; f < 4; ++f) {
      v2f bb = *(const v2f*)(brow[f] + 4 * k);
      acc[f] = __builtin_amdgcn_wmma_f32_16x16x4_f32(
          false, a, false, bb, (short)0, acc[f], false, false);
    }
  }

  // ----- fused masked max epilogue -----
  int iBase = i0 + 8 * half;                    // rows owned by this lane half
  int ip2r[8], ip2c[8];
#pragma unroll
  for (int r = 0; r < 8; ++r) {
    ip2r[r] = P2[2 * (iBase + r)];
    ip2c[r] = P2[2 * (iBase + r) + 1];
  }
  float rowm[8];
#pragma unroll
  for (int r = 0; r < 8; ++r) rowm[r] = -1.0e30f;

#pragma unroll
  for (int f = 0; f < 4; ++f) {
    int j    = tJ + 16 * f + l;                 // column owned by this lane
    int jp2r = P2[2 * j], jp2c = P2[2 * j + 1];
    int jgr  = j >> 6,    jgc  = j & 63;
    float colm = -1.0e30f;
#pragma unroll
    for (int r = 0; r < 8; ++r) {
      int   i = iBase + r;
      float s = acc[f][r];
      int dr2 = abs(ip2r[r] - jp2r), dc2 = abs(ip2c[r] - jp2c);
      float v2v = s - ((max(dr2, dc2) <= 4) ? 5.0f : 0.0f);  // image-2 mask
      int dr1 = abs((i >> 6) - jgr), dc1 = abs((i & 63) - jgc);
      float v1v = s - ((max(dr1, dc1) <= 4) ? 5.0f : 0.0f);  // image-1 grid mask
      colm    = fmaxf(colm, v1v);
      rowm[r] = fmaxf(rowm[r], v2v);
    }
    // lanes l and l+16 share column j: combine halves, one atomic per column
    colm = fmaxf(colm, __shfl_xor(colm, 16, 32));
    if (half == 0) atomicMaxF(&M1[(size_t)b * N_ + j], colm);
  }
  // row max: butterfly within each 16-lane half (xor 1/2/4/8 stays in-half)
#pragma unroll
  for (int r = 0; r < 8; ++r) {
    float v = rowm[r];
    v = fmaxf(v, __shfl_xor(v, 1, 32));
    v = fmaxf(v, __shfl_xor(v, 2, 32));
    v = fmaxf(v, __shfl_xor(v, 4, 32));
    v = fmaxf(v, __shfl_xor(v, 8, 32));
    if (l == r) atomicMaxF(&M2[(size_t)b * N_ + iBase + r], v);
  }
}

// ---------------------------------------------------------------------------
// finalize: positives from descriptor dots, neg = 2-2*max(M1,M2), hinge loss,
// hierarchical reduction -> 3 atomicAdds into d_out = (loss, pos, neg) means.
// ---------------------------------------------------------------------------
__global__ void __launch_bounds__(256)
finalize_kernel(const float* __restrict__ d1T, const float* __restrict__ d2T,
                const float* __restrict__ M1, const float* __restrict__ M2,
                float* __restrict__ out) {
  int idx = blockIdx.x * 256 + threadIdx.x;     // exactly B_*N_ threads
  const float* r1 = d1T + (size_t)idx * C_;
  const float* r2 = d2T + (size_t)idx * C_;
  float dot = 0.0f;
#pragma unroll 8
  for (int c = 0; c < C_; ++c) dot += r1[c] * r2[c];
  float pos  = 2.0f - 2.0f * dot;
  float neg  = 2.0f - 2.0f * fmaxf(M1[idx], M2[idx]);
  float loss = fmaxf(0.0f, 1.0f + pos - neg);   // MARGIN = 1

  for (int off = 16; off > 0; off >>= 1) {
    loss += __shfl_down(loss, off, 32);
    pos  += __shfl_down(pos,  off, 32);
    neg  += __shfl_down(neg,  off, 32);
  }
  __shared__ float sl[8], sp[8], sn[8];
  int w = threadIdx.x >> 5;
  if ((threadIdx.x & 31) == 0) { sl[w] = loss; sp[w] = pos; sn[w] = neg; }
  __syncthreads();
  if (threadIdx.x == 0) {
    float L = 0.f, P = 0.f, G = 0.f;
    for (int i = 0; i < 8; ++i) { L += sl[i]; P += sp[i]; G += sn[i]; }
    const float s = 1.0f / (float)(N_ * B_);    // mean over N, then /B
    atomicAdd(&out[0], L * s);
    atomicAdd(&out[1], P * s);
    atomicAdd(&out[2], G * s);
  }
}

// ---------------------------------------------------------------------------
extern "C" void kernel_launch(void* const* d_in, const int* in_sizes, int n_in,
                              void* d_out, int out_size, void* d_ws, size_t ws_size,
                              hipStream_t stream) {
  const float* f1 = (const float*)d_in[0];  // img1_1 [B,C,H,W]
  const float* f2 = (const float*)d_in[1];  // img1_2 [B,C,H,W]
  const float* tc = (const float*)d_in[2];  // [B,2,N]
  float* out = (float*)d_out;

  char* ws = (char*)d_ws;
  float* d1T  = (float*)(ws + OFF_D1T);
  float* d2T  = (float*)(ws + OFF_D2T);
  int*   pos2 = (int*)  (ws + OFF_POS);
  float* M1   = (float*)(ws + OFF_M1);
  float* M2   = (float*)(ws + OFF_M2);

  init_kernel<<<dim3((B_ * N_ + 255) / 256), dim3(256), 0, stream>>>(tc, pos2, M1, M2, out);
  prep_desc<<<dim3(N_ / 64, B_), dim3(64), 0, stream>>>(f1, pos2, 0, d1T);
  prep_desc<<<dim3(N_ / 64, B_), dim3(64), 0, stream>>>(f2, pos2, 1, d2T);
  gemm_minmax_kernel<<<dim3((N_ / TILE_) * (N_ / TILE_), B_), dim3(128), 0, stream>>>(
      d1T, d2T, pos2, M1, M2);
  finalize_kernel<<<dim3(B_ * N_ / 256), dim3(256), 0, stream>>>(d1T, d2T, M1, M2, out);
}